// SeperationBlock_Stateful_49916109914573
// MI455X (gfx1250) — compile-verified
//
#include <hip/hip_runtime.h>
#include <hip/hip_bf16.h>

// ---------------------------------------------------------------------------
// Problem constants (from the reference)
// ---------------------------------------------------------------------------
#define NBATCH 64
#define TSTEPS 1000
#define DIN    257
#define HID    128
#define GATES  512           // 4*HID
#define HPAD   132           // LDS row stride (128 + 4 pad) -> bank-conflict free
#define SLAB   (NBATCH*HPAD) // 8448 floats per staging slab

typedef float v2f __attribute__((ext_vector_type(2)));
typedef float v8f __attribute__((ext_vector_type(8)));
typedef unsigned int v4u __attribute__((ext_vector_type(4)));
typedef int v8i __attribute__((ext_vector_type(8)));
typedef int v4i __attribute__((ext_vector_type(4)));

static __device__ __forceinline__ v8f wmma4(v2f a, v2f b, v8f c) {
    // D = A(16x4 f32) x B(4x16 f32) + C(16x16 f32)
    return __builtin_amdgcn_wmma_f32_16x16x4_f32(false, a, false, b, (short)0, c, false, false);
}

static __device__ __forceinline__ float sigf(float x) {
    return 1.f / (1.f + __expf(-x));
}
static __device__ __forceinline__ float tanhf_(float x) {
    float e = __expf(-2.f * x);
    return (1.f - e) / (1.f + e);
}

// ---------------------------------------------------------------------------
// TDM: async 3D tile load  pre1[t][0:64][j*32 + gate*128 + 0:32] -> LDS slab
// tile_dim = (32, 4, 64), strides (128, 512), data_size 4B,
// pad 4 DWORDs every 128 DWORDs -> LDS rows land at stride HPAD=132.
// (6-arg clang-23 builtin: g0, g1, g2, g3, extra group, cpol)
// ---------------------------------------------------------------------------
static __device__ __forceinline__ void tdm_issue(const float* gsrc, unsigned ldsOff) {
    unsigned long long ga = (unsigned long long)(uintptr_t)gsrc;
    v4u g0;
    g0[0] = 1u;                                  // count=1, user descriptor
    g0[1] = ldsOff;                              // lds_addr (bytes)
    g0[2] = (unsigned)ga;                        // global_addr[31:0]
    g0[3] = (unsigned)((ga >> 32) & 0x01FFFFFFull) | (2u << 30);  // addr hi | type=2
    v8i g1;
    g1[0] = (int)((2u << 16)      // data_size = 4B
                | (1u << 20)      // pad_enable
                | (6u << 22)      // pad_interval: 128 DWORDs
                | (3u << 25));    // pad_amount: 4 DWORDs
    g1[1] = (int)(0x7FFFu << 16); // tensor_dim0 lo (generous, never OOB)
    g1[2] = (int)(0x7FFFu << 16); // tensor_dim0 hi=0 | tensor_dim1 lo
    g1[3] = (int)(32u << 16);     // tensor_dim1 hi=0 | tile_dim0 = 32
    g1[4] = (int)(4u | (64u << 16)); // tile_dim1 = 4 | tile_dim2 = 64
    g1[5] = 128;                  // tensor_dim0_stride lo32 = 128
    g1[6] = (int)(512u << 16);    // stride0 hi=0 | tensor_dim1_stride lo16 = 512
    g1[7] = 0;                    // tensor_dim1_stride hi = 0
    v4i g2; g2[0] = 0x7FFF; g2[1] = 0; g2[2] = 0; g2[3] = 0;  // tensor_dim2 big
    v4i g3; g3[0] = 0; g3[1] = 0; g3[2] = 0; g3[3] = 0;
    v8i gx; gx[0]=0; gx[1]=0; gx[2]=0; gx[3]=0; gx[4]=0; gx[5]=0; gx[6]=0; gx[7]=0;
    __builtin_amdgcn_tensor_load_to_lds(g0, g1, g2, g3, gx, 0);
}

// Device-scope split barrier across the 8 persistent workgroups.
static __device__ __forceinline__ void sys_barrier(unsigned* slot, unsigned nwg) {
    __syncthreads();
    __threadfence();
    if (threadIdx.x == 0) {
        __hip_atomic_fetch_add(slot, 1u, __ATOMIC_RELEASE, __HIP_MEMORY_SCOPE_AGENT);
        while (__hip_atomic_load(slot, __ATOMIC_ACQUIRE, __HIP_MEMORY_SCOPE_AGENT) < nwg) {
            __builtin_amdgcn_s_sleep(2);
        }
    }
    __syncthreads();
}

// ---------------------------------------------------------------------------
// Kernel 0: zero the barrier counters (graph-replay determinism).
// ---------------------------------------------------------------------------
__global__ void init_cnt_kernel(unsigned* __restrict__ cnt) {
    int i = blockIdx.x * blockDim.x + threadIdx.x;
    if (i < TSTEPS + 2) cnt[i] = 0u;
}

// ---------------------------------------------------------------------------
// Kernel 1: pre1[t*64+n][g] = sum_k x[n][t][k]*Wih1[g][k] + bih1[g]+bhh1[g]
// Block: 256 threads (8 waves), 32 output rows x 512 cols.
// Wave: 2 m-tiles x 4 n-tiles of WMMA f32 16x16x4.
// ---------------------------------------------------------------------------
__global__ void __launch_bounds__(256) pre1_kernel(
        const float* __restrict__ x, const float* __restrict__ Wih,
        const float* __restrict__ bih, const float* __restrict__ bhh,
        float* __restrict__ pre) {
    const int wave = threadIdx.x >> 5;
    const int lane = threadIdx.x & 31;
    const int lo = lane & 15, hi = lane >> 4;
    const int m0 = blockIdx.x * 32;
    const int n0 = wave * 64;

    long arow[2];
    #pragma unroll
    for (int mt = 0; mt < 2; ++mt) {
        int r = m0 + mt * 16 + lo;
        int nb = r & 63, tt = r >> 6;
        arow[mt] = (long)nb * (TSTEPS * DIN) + (long)tt * DIN;
    }
    long brow[4];
    #pragma unroll
    for (int nt = 0; nt < 4; ++nt) brow[nt] = (long)(n0 + nt * 16 + lo) * DIN;

    v8f acc[2][4];
    #pragma unroll
    for (int mt = 0; mt < 2; ++mt)
        #pragma unroll
        for (int nt = 0; nt < 4; ++nt)
            #pragma unroll
            for (int rr = 0; rr < 8; ++rr) acc[mt][nt][rr] = 0.f;

    // 64 full K-steps of 4 (k = 0..255); rows are 4B-aligned only -> scalar pair loads
    for (int kk = 0; kk < 64; ++kk) {
        const int k = kk * 4 + 2 * hi;
        v2f a[2], b[4];
        #pragma unroll
        for (int mt = 0; mt < 2; ++mt) { a[mt][0] = x[arow[mt] + k]; a[mt][1] = x[arow[mt] + k + 1]; }
        #pragma unroll
        for (int nt = 0; nt < 4; ++nt) { b[nt][0] = Wih[brow[nt] + k]; b[nt][1] = Wih[brow[nt] + k + 1]; }
        #pragma unroll
        for (int mt = 0; mt < 2; ++mt)
            #pragma unroll
            for (int nt = 0; nt < 4; ++nt)
                acc[mt][nt] = wmma4(a[mt], b[nt], acc[mt][nt]);
    }
    // K tail: k = 256 (only low half-wave, first element valid)
    {
        v2f a[2], b[4];
        #pragma unroll
        for (int mt = 0; mt < 2; ++mt) { float v = x[arow[mt] + 256]; a[mt][0] = hi ? 0.f : v; a[mt][1] = 0.f; }
        #pragma unroll
        for (int nt = 0; nt < 4; ++nt) { float v = Wih[brow[nt] + 256]; b[nt][0] = hi ? 0.f : v; b[nt][1] = 0.f; }
        #pragma unroll
        for (int mt = 0; mt < 2; ++mt)
            #pragma unroll
            for (int nt = 0; nt < 4; ++nt)
                acc[mt][nt] = wmma4(a[mt], b[nt], acc[mt][nt]);
    }
    // Store with folded biases (bih + bhh)
    #pragma unroll
    for (int nt = 0; nt < 4; ++nt) {
        const int col = n0 + nt * 16 + lo;
        const float bs = bih[col] + bhh[col];
        #pragma unroll
        for (int mt = 0; mt < 2; ++mt)
            #pragma unroll
            for (int rr = 0; rr < 8; ++rr) {
                const int r = m0 + mt * 16 + rr + 8 * hi;
                pre[(long)r * GATES + col] = acc[mt][nt][rr] + bs;
            }
    }
}

// ---------------------------------------------------------------------------
// Kernel 2: persistent two-layer LSTM recurrence.
// 8 workgroups x 256 threads. WGs 0-3: layer 1; WGs 4-7: layer 2 (one step
// behind, so a single device barrier per iteration suffices).
// Each WG owns hidden slice [j*32, j*32+32) => 128 gate rows; recurrent
// weight B-fragments live in VGPRs for the whole sequence.
// ---------------------------------------------------------------------------
__global__ void __launch_bounds__(256) lstm_kernel(
        const float* __restrict__ inStates,
        const float* __restrict__ Whh1,
        const float* __restrict__ Wih2, const float* __restrict__ Whh2,
        const float* __restrict__ bih2, const float* __restrict__ bhh2,
        const float* __restrict__ pre1,
        float* __restrict__ x2seq,
        float* __restrict__ h1pub, float* __restrict__ h2pub,
        unsigned* __restrict__ cnt,
        float* __restrict__ outStates) {
    extern __shared__ float smem[];
    float* slab0 = smem;             // L1: pre ping  | L2: h1 stage
    float* slab1 = smem + SLAB;      // L1: pre pong  | L2: h2 stage
    float* slab2 = smem + 2 * SLAB;  // L1: h1 stage  | L2: unused
    float* gst   = smem + 3 * SLAB;  // gates [64][HPAD]
    float* cst   = smem + 4 * SLAB;  // cell state [64][32]

    const int tid = threadIdx.x;
    const int wave = tid >> 5, lane = tid & 31, lo = lane & 15, hi = lane >> 4;
    const int wg = blockIdx.x;
    const bool isL2 = wg >= 4;
    const int j = isL2 ? wg - 4 : wg;

    // Lane's owned output column within the local 128-gate slice.
    const int ln = wave * 16 + lo;             // 0..127
    const int gate = ln >> 5, hh5 = ln & 31;
    const int grow = gate * HID + j * 32 + hh5; // row in [4H, H] weight matrix

    // --- Preload recurrent-weight B fragments into registers (8B-aligned) ---
    v2f B1[32], B2[32];
    if (!isL2) {
        #pragma unroll
        for (int kk = 0; kk < 32; ++kk)
            B1[kk] = *reinterpret_cast<const v2f*>(Whh1 + (long)grow * HID + 4 * kk + 2 * hi);
    } else {
        #pragma unroll
        for (int kk = 0; kk < 32; ++kk) {
            B1[kk] = *reinterpret_cast<const v2f*>(Wih2 + (long)grow * HID + 4 * kk + 2 * hi);
            B2[kk] = *reinterpret_cast<const v2f*>(Whh2 + (long)grow * HID + 4 * kk + 2 * hi);
        }
    }
    const float b2sum = isL2 ? (bih2[grow] + bhh2[grow]) : 0.f;

    // --- Init cell state (in_states rows: h1=0..63, c1=64..127, h2=128..191, c2=192..255) ---
    const int cchunk = isL2 ? 192 : 64;
    for (int p = 0; p < 8; ++p) {
        int idx = tid + p * 256;                 // 0..2047
        int b = idx >> 5, h5 = idx & 31;
        cst[idx] = inStates[(long)(cchunk + b) * HID + j * 32 + h5];
    }
    // --- Publish initial h into slot 0 of the rolling buffers ---
    if (wg == 0) for (int p = 0; p < 32; ++p) { int e = tid + p * 256; h1pub[e] = inStates[e]; }
    if (wg == 4) for (int p = 0; p < 32; ++p) { int e = tid + p * 256; h2pub[e] = inStates[2 * NBATCH * HID + e]; }

    // --- Pre-issue TDM for t=0 into slab0 ---
    if (!isL2 && wave == 0)
        tdm_issue(pre1 + (long)j * 32, (unsigned)(uintptr_t)slab0);

    sys_barrier(&cnt[0], 8u);

    float* oH = outStates + (isL2 ? 2 * NBATCH * HID : 0);
    float* oC = outStates + (isL2 ? 3 * NBATCH * HID : 1 * NBATCH * HID);

    for (int it = 0; it <= TSTEPS; ++it) {
        if (!isL2) {
            const int t = it;
            if (t < TSTEPS) {
                // Prefetch next pre1 slice via TDM, then wait for the current one.
                if (wave == 0) {
                    if (t + 1 < TSTEPS) {
                        float* dst = ((t + 1) & 1) ? slab1 : slab0;
                        tdm_issue(pre1 + (long)(t + 1) * NBATCH * GATES + j * 32,
                                  (unsigned)(uintptr_t)dst);
                        __builtin_amdgcn_s_wait_tensorcnt(1);  // in-order: slot t done
                    } else {
                        __builtin_amdgcn_s_wait_tensorcnt(0);
                    }
                }
                // Stage h1[t-1] from L2-resident publish buffer into LDS (padded rows).
                const float* hsrc = h1pub + (long)(t & 1) * (NBATCH * HID);
                for (int p = 0; p < 8; ++p) {
                    int q = tid + p * 256;            // float4 index, 0..2047
                    int b = q >> 5, k4 = (q & 31) * 4;
                    float4 v = reinterpret_cast<const float4*>(hsrc)[q];
                    *reinterpret_cast<float4*>(slab2 + b * HPAD + k4) = v;
                }
                __syncthreads();

                // gates = pre1_slice + h1 @ Whh1_ownᵀ   (M=64, N=16/wave, K=128)
                const float* pslab = (t & 1) ? slab1 : slab0;
                #pragma unroll
                for (int mt = 0; mt < 4; ++mt) {
                    v8f acc;
                    #pragma unroll
                    for (int rr = 0; rr < 8; ++rr)
                        acc[rr] = pslab[(mt * 16 + rr + 8 * hi) * HPAD + ln];
                    #pragma unroll
                    for (int kk = 0; kk < 32; ++kk) {
                        v2f a = *reinterpret_cast<const v2f*>(
                            slab2 + (mt * 16 + lo) * HPAD + 4 * kk + 2 * hi);
                        acc = wmma4(a, B1[kk], acc);
                    }
                    #pragma unroll
                    for (int rr = 0; rr < 8; ++rr)
                        gst[(mt * 16 + rr + 8 * hi) * HPAD + ln] = acc[rr];
                }
                __syncthreads();

                // Elementwise gate math; publish h1[t] into the other slot.
                float* hdst = h1pub + (long)((t + 1) & 1) * (NBATCH * HID);
                for (int p = 0; p < 8; ++p) {
                    int idx = tid + p * 256;
                    int b = idx >> 5, h5 = idx & 31;
                    float gi = gst[b * HPAD + h5];
                    float gf = gst[b * HPAD + 32 + h5];
                    float gg = gst[b * HPAD + 64 + h5];
                    float go = gst[b * HPAD + 96 + h5];
                    float c = sigf(gf) * cst[idx] + sigf(gi) * tanhf_(gg);
                    float h = sigf(go) * tanhf_(c);
                    cst[idx] = c;
                    hdst[(long)b * HID + j * 32 + h5] = h;
                    oH[(long)b * HID + j * 32 + h5] = h;
                    oC[(long)b * HID + j * 32 + h5] = c;
                }
            }
        } else {
            const int t2 = it - 1;   // layer 2 runs one step behind
            if (t2 >= 0) {
                const float* s1 = h1pub + (long)((t2 + 1) & 1) * (NBATCH * HID); // h1[t2]
                const float* s2 = h2pub + (long)(t2 & 1) * (NBATCH * HID);       // h2[t2-1]
                for (int p = 0; p < 8; ++p) {
                    int q = tid + p * 256;
                    int b = q >> 5, k4 = (q & 31) * 4;
                    *reinterpret_cast<float4*>(slab0 + b * HPAD + k4) =
                        reinterpret_cast<const float4*>(s1)[q];
                    *reinterpret_cast<float4*>(slab1 + b * HPAD + k4) =
                        reinterpret_cast<const float4*>(s2)[q];
                }
                __syncthreads();

                // gates = bias + h1[t2] @ Wih2ᵀ + h2[t2-1] @ Whh2ᵀ
                #pragma unroll
                for (int mt = 0; mt < 4; ++mt) {
                    v8f acc;
                    #pragma unroll
                    for (int rr = 0; rr < 8; ++rr) acc[rr] = b2sum;
                    #pragma unroll
                    for (int kk = 0; kk < 32; ++kk) {
                        v2f a1 = *reinterpret_cast<const v2f*>(
                            slab0 + (mt * 16 + lo) * HPAD + 4 * kk + 2 * hi);
                        acc = wmma4(a1, B1[kk], acc);
                        v2f a2 = *reinterpret_cast<const v2f*>(
                            slab1 + (mt * 16 + lo) * HPAD + 4 * kk + 2 * hi);
                        acc = wmma4(a2, B2[kk], acc);
                    }
                    #pragma unroll
                    for (int rr = 0; rr < 8; ++rr)
                        gst[(mt * 16 + rr + 8 * hi) * HPAD + ln] = acc[rr];
                }
                __syncthreads();

                float* hdst = h2pub + (long)((t2 + 1) & 1) * (NBATCH * HID);
                float* xdst = x2seq + (long)t2 * (NBATCH * HID);
                for (int p = 0; p < 8; ++p) {
                    int idx = tid + p * 256;
                    int b = idx >> 5, h5 = idx & 31;
                    float gi = gst[b * HPAD + h5];
                    float gf = gst[b * HPAD + 32 + h5];
                    float gg = gst[b * HPAD + 64 + h5];
                    float go = gst[b * HPAD + 96 + h5];
                    float c = sigf(gf) * cst[idx] + sigf(gi) * tanhf_(gg);
                    float h = sigf(go) * tanhf_(c);
                    cst[idx] = c;
                    hdst[(long)b * HID + j * 32 + h5] = h;
                    xdst[(long)b * HID + j * 32 + h5] = h;
                    oH[(long)b * HID + j * 32 + h5] = h;
                    oC[(long)b * HID + j * 32 + h5] = c;
                }
            }
        }
        sys_barrier(&cnt[1 + it], 8u);
    }
}

// ---------------------------------------------------------------------------
// Kernel 3: mask[n][t][d] = sigmoid( x2[t][n][:] . Wd[d][:] + bd[d] )
// One wave per 16x16 output tile; K=128 in 32 WMMA f32 16x16x4 steps.
// ---------------------------------------------------------------------------
__global__ void __launch_bounds__(32) mask_kernel(
        const float* __restrict__ x2, const float* __restrict__ Wd,
        const float* __restrict__ bd, float* __restrict__ out) {
    const int lane = threadIdx.x & 31;
    const int lo = lane & 15, hi = lane >> 4;
    const int m0 = blockIdx.x * 16;
    const int col = blockIdx.y * 16 + lo;
    const bool cv = col < DIN;
    const float msk = cv ? 1.f : 0.f;

    const float* arow = x2 + (long)(m0 + lo) * HID;
    const float* brow = Wd + (long)(cv ? col : 0) * HID;

    v8f acc;
    #pragma unroll
    for (int rr = 0; rr < 8; ++rr) acc[rr] = 0.f;

    #pragma unroll
    for (int kk = 0; kk < 32; ++kk) {
        v2f a = *reinterpret_cast<const v2f*>(arow + 4 * kk + 2 * hi);
        v2f b = *reinterpret_cast<const v2f*>(brow + 4 * kk + 2 * hi);
        b *= msk;
        acc = wmma4(a, b, acc);
    }
    const float bb = cv ? bd[col] : 0.f;
    #pragma unroll
    for (int rr = 0; rr < 8; ++rr) {
        const int r = m0 + rr + 8 * hi;
        const int tt = r >> 6, nb = r & 63;
        if (cv)
            out[(long)nb * (TSTEPS * DIN) + (long)tt * DIN + col] = sigf(acc[rr] + bb);
    }
}

// ---------------------------------------------------------------------------
// Host launcher
// ---------------------------------------------------------------------------
extern "C" void kernel_launch(void* const* d_in, const int* in_sizes, int n_in,
                              void* d_out, int out_size, void* d_ws, size_t ws_size,
                              hipStream_t stream) {
    const float* x        = (const float*)d_in[0];
    const float* inStates = (const float*)d_in[1];
    const float* Wih1     = (const float*)d_in[2];
    const float* Whh1     = (const float*)d_in[3];
    const float* bih1     = (const float*)d_in[4];
    const float* bhh1     = (const float*)d_in[5];
    const float* Wih2     = (const float*)d_in[6];
    const float* Whh2     = (const float*)d_in[7];
    const float* bih2     = (const float*)d_in[8];
    const float* bhh2     = (const float*)d_in[9];
    const float* Wd       = (const float*)d_in[10];
    const float* bd       = (const float*)d_in[11];

    float* mask = (float*)d_out;                                   // [64,1000,257]
    float* outStates = mask + (size_t)NBATCH * TSTEPS * DIN;       // [4*64,128]

    // Workspace carve-up
    float* wsf   = (float*)d_ws;
    float* pre1  = wsf;                                            // [T][64][512]
    float* x2seq = pre1 + (size_t)TSTEPS * NBATCH * GATES;         // [T][64][128]
    float* h1pub = x2seq + (size_t)TSTEPS * NBATCH * HID;          // [2][64][128]
    float* h2pub = h1pub + 2 * NBATCH * HID;                       // [2][64][128]
    unsigned* cnt = (unsigned*)(h2pub + 2 * NBATCH * HID);         // [T+2]

    const int lstmSmem = (4 * SLAB + NBATCH * 32) * (int)sizeof(float); // 143360 B
    (void)hipFuncSetAttribute((const void*)lstm_kernel,
                              hipFuncAttributeMaxDynamicSharedMemorySize, lstmSmem);

    // 0) zero barrier counters (deterministic across graph replays)
    init_cnt_kernel<<<8, 256, 0, stream>>>(cnt);
    // 1) layer-1 input projection + fused biases (WMMA f32, HBM-bound)
    pre1_kernel<<<(NBATCH * TSTEPS) / 32, 256, 0, stream>>>(x, Wih1, bih1, bhh1, pre1);
    // 2) persistent recurrence: 8 cooperating WGs, TDM-fed, weights in VGPRs
    lstm_kernel<<<8, 256, lstmSmem, stream>>>(inStates, Whh1, Wih2, Whh2, bih2, bhh2,
                                              pre1, x2seq, h1pub, h2pub, cnt, outStates);
    // 3) dense sigmoid head (WMMA f32, HBM-bound)
    dim3 mg((NBATCH * TSTEPS) / 16, (DIN + 15) / 16);
    mask_kernel<<<mg, 32, 0, stream>>>(x2seq, Wd, bd, mask);
}